// NTMHead_74028056313978
// MI455X (gfx1250) — compile-verified
//
#include <hip/hip_runtime.h>
#include <math.h>

#define B     64
#define CTRL  512
#define KEY   64
#define MEMU  16384

typedef float v2f __attribute__((ext_vector_type(2)));
typedef float v8f __attribute__((ext_vector_type(8)));
typedef int   v4i_t __attribute__((ext_vector_type(4)));

// ---- workspace layout (float offsets) ----
#define WS_K      0        // B*KEY
#define WS_KNORM  4096     // B
#define WS_BETA   4160     // B
#define WS_G      4224     // B
#define WS_Y      4288     // B
#define WS_S      4352     // 3*B
#define WS_SUMEXP 4544     // B
#define WS_SUMSH  4608     // B
#define WS_BUF0   8192     // B*MEMU (w_i buffer)

__device__ __forceinline__ float softplus_f(float x) {
    return (x > 20.f) ? x : log1pf(expf(x));
}

// ---------------------------------------------------------------------------
// gfx1250 async copy: global -> LDS, 16 bytes per lane, tracked by ASYNCcnt.
// The builtin's signature (from the compiler diagnostic) takes int32x4
// pointers: (global int4*, shared int4*, imm offset, imm cpol).
// ---------------------------------------------------------------------------
__device__ __forceinline__ void async_copy_b128(const float* g, float* l) {
#if __has_builtin(__builtin_amdgcn_global_load_async_to_lds_b128)
    typedef __attribute__((address_space(1))) v4i_t* gp_t;
    typedef __attribute__((address_space(3))) v4i_t* lp_t;
    __builtin_amdgcn_global_load_async_to_lds_b128(
        (gp_t)(v4i_t*)(float*)g,
        (lp_t)(v4i_t*)l,
        0, 0);
#else
    unsigned int lds_off = (unsigned int)(unsigned long long)l;
    asm volatile("global_load_async_to_lds_b128 %0, %1, off"
                 :: "v"(lds_off), "v"((unsigned long long)g)
                 : "memory");
#endif
}

__device__ __forceinline__ void wait_async0() {
#if __has_builtin(__builtin_amdgcn_s_wait_asynccnt)
    __builtin_amdgcn_s_wait_asynccnt(0);
#else
    asm volatile("s_wait_asynccnt 0x0" ::: "memory");
#endif
}

// ---------------------------------------------------------------------------
// Kernel 1: tiny linears from controller state -> k, beta, g, y, s, ||k||.
// Also zeroes the per-batch accumulators and the `data` output region
// (must happen every launch; harness does not re-poison between replays).
// grid = B blocks, 256 threads.
// ---------------------------------------------------------------------------
__global__ void head_params(const float* __restrict__ cs,
                            const float* __restrict__ key_W, const float* __restrict__ key_b,
                            const float* __restrict__ ks_W,  const float* __restrict__ ks_b,
                            const float* __restrict__ ig_W,  const float* __restrict__ ig_b,
                            const float* __restrict__ sw_W,  const float* __restrict__ sw_b,
                            const float* __restrict__ sf_W,  const float* __restrict__ sf_b,
                            float* __restrict__ ws, float* __restrict__ dout)
{
    __shared__ float red[256];
    __shared__ float klds[KEY];
    __shared__ float scal[8];
    const int b = blockIdx.x;
    const int t = threadIdx.x;
    const float* csb = cs + (size_t)b * CTRL;

    // ---- k = cs @ key_W.T + key_b : 64 outputs, 4 threads each ----
    {
        const int j = t >> 2, part = t & 3;
        const float* wrow = key_W + (size_t)j * CTRL + part * 128;
        const float* csp  = csb + part * 128;
        float acc = 0.f;
        for (int c = 0; c < 128; ++c) acc = fmaf(csp[c], wrow[c], acc);
        red[t] = acc;
        __syncthreads();
        if (part == 0) {
            float kv = red[t] + red[t + 1] + red[t + 2] + red[t + 3] + key_b[j];
            ws[WS_K + b * KEY + j] = kv;
            klds[j] = kv;
        }
        __syncthreads();
    }
    // ---- ||k|| ----
    red[t] = (t < KEY) ? klds[t] * klds[t] : 0.f;
    __syncthreads();
    for (int s = 128; s > 0; s >>= 1) { if (t < s) red[t] += red[t + s]; __syncthreads(); }
    if (t == 0) ws[WS_KNORM + b] = sqrtf(red[0]);

    // ---- six length-512 dots: ks, ig, sf, sw0, sw1, sw2 ----
    for (int r = 0; r < 6; ++r) {
        const float* wr = (r == 0) ? ks_W : (r == 1) ? ig_W : (r == 2) ? sf_W
                          : sw_W + (size_t)(r - 3) * CTRL;
        float p = fmaf(csb[t], wr[t], csb[t + 256] * wr[t + 256]);
        red[t] = p;
        __syncthreads();
        for (int s = 128; s > 0; s >>= 1) { if (t < s) red[t] += red[t + s]; __syncthreads(); }
        if (t == 0) scal[r] = red[0];
        __syncthreads();
    }
    if (t == 0) {
        float beta = softplus_f(scal[0] + ks_b[0]);
        float g    = 1.f / (1.f + expf(-(scal[1] + ig_b[0])));
        float y    = 1.f + softplus_f(scal[2] + sf_b[0]);
        float l0 = scal[3] + sw_b[0], l1 = scal[4] + sw_b[1], l2 = scal[5] + sw_b[2];
        float mx = fmaxf(l0, fmaxf(l1, l2));
        float e0 = expf(l0 - mx), e1 = expf(l1 - mx), e2 = expf(l2 - mx);
        float inv = 1.f / (e0 + e1 + e2);
        ws[WS_BETA + b] = beta;
        ws[WS_G + b]    = g;
        ws[WS_Y + b]    = y;
        ws[WS_S + 3 * b + 0] = e0 * inv;
        ws[WS_S + 3 * b + 1] = e1 * inv;
        ws[WS_S + 3 * b + 2] = e2 * inv;
        ws[WS_SUMEXP + b] = 0.f;
        ws[WS_SUMSH + b]  = 0.f;
    }
    if (t < KEY) dout[(size_t)B * MEMU + b * KEY + t] = 0.f;   // zero `data` region
}

// ---------------------------------------------------------------------------
// Kernel 2: first full pass over memory. Tile staged global->LDS with
// GLOBAL_LOAD_ASYNC_TO_LDS_B128 (per-lane LDS scatter into padded stride).
// Per wave: 16 memory rows; dots via a chain of 16 V_WMMA_F32_16X16X4_F32
// (k broadcast across all 16 B-columns), row sum-of-squares via VALU; then
// score -> e = exp(score - beta) and a block-reduced atomicAdd of the
// softmax partition sum.
// grid = B * (MEMU/128) blocks, 256 threads (8 waves * 16 rows = 128 rows).
// ---------------------------------------------------------------------------
#define ROWS 128
#define LDST 68   // padded LDS row stride (floats) -> conflict-free fragment reads

__global__ void dots_kernel(const float* __restrict__ mem,
                            const float* __restrict__ ws,
                            float* __restrict__ ereg,     // d_out w-region (holds e)
                            float* __restrict__ sumexp)
{
    __shared__ __align__(16) float A[ROWS * LDST];
    __shared__ float kl[KEY];
    __shared__ float dotsL[ROWS];
    __shared__ float ssL[ROWS];
    __shared__ float red[256];

    const int t    = threadIdx.x;
    const int b    = blockIdx.x >> 7;     // / (MEMU/ROWS) = /128
    const int tile = blockIdx.x & 127;
    const int m0   = tile * ROWS;
    const float* mbase = mem + ((size_t)b * MEMU + m0) * KEY;

    // async stage: 128 rows x 64 floats -> LDS, 16B per lane, padded stride.
    #pragma unroll
    for (int i = 0; i < 8; ++i) {
        int idx4 = t + i * 256;          // 0..2047 float4s
        int row  = idx4 >> 4;
        int c4   = idx4 & 15;
        async_copy_b128(mbase + (size_t)idx4 * 4, &A[row * LDST + c4 * 4]);
    }
    if (t < KEY) kl[t] = ws[WS_K + b * KEY + t];
    wait_async0();
    __syncthreads();

    // WMMA dot chain over K=64 in chunks of 4
    const int lane = t & 31;
    const int wv   = t >> 5;
    const int hi   = lane >> 4;          // 0: K%4 in {0,1}; 1: K%4 in {2,3}
    const int mrow = lane & 15;
    const int rowBase = wv * 16;
    const float* arow = &A[(rowBase + mrow) * LDST];

    v8f c8 = {};
    float ss = 0.f;
    #pragma unroll
    for (int kk = 0; kk < 16; ++kk) {
        const int coff = kk * 4 + hi * 2;
        float a0 = arow[coff], a1 = arow[coff + 1];
        float b0 = kl[coff],   b1 = kl[coff + 1];
        ss = fmaf(a0, a0, ss);
        ss = fmaf(a1, a1, ss);
        v2f av = {a0, a1};
        v2f bv = {b0, b1};
        c8 = __builtin_amdgcn_wmma_f32_16x16x4_f32(false, av, false, bv,
                                                   (short)0, c8, false, false);
    }
    // combine the two K-halves of the row sumsq
    ss += __shfl_xor(ss, 16, 32);
    if (!hi) ssL[rowBase + mrow] = ss;
    // C layout: VGPR r, lanes 0-15 -> M=r ; lanes 16-31 -> M=8+r (any N column)
    if (mrow == 0) {
        #pragma unroll
        for (int r = 0; r < 8; ++r) dotsL[rowBase + hi * 8 + r] = c8[r];
    }
    __syncthreads();

    const float beta = ws[WS_BETA + b];
    const float kn   = ws[WS_KNORM + b];
    float e = 0.f;
    if (t < ROWS) {
        float denom = fmaxf(sqrtf(ssL[t]) * kn, 1e-8f);
        float score = beta * dotsL[t] / denom;
        e = expf(score - beta);          // score <= beta -> never overflows
        ereg[(size_t)b * MEMU + m0 + t] = e;
    }
    red[t] = e;
    __syncthreads();
    for (int s = 128; s > 0; s >>= 1) { if (t < s) red[t] += red[t + s]; __syncthreads(); }
    if (t == 0) atomicAdd(&sumexp[b], red[0]);
}

// ---------------------------------------------------------------------------
// Kernel 3: w_c = e/sum ; w_i = g*w_c + (1-g)*prev_w   (elementwise)
// ---------------------------------------------------------------------------
__global__ void interp_kernel(const float* __restrict__ prev_w,
                              const float* __restrict__ ws,
                              const float* __restrict__ ereg,
                              float* __restrict__ wi)
{
    const int idx = blockIdx.x * 256 + threadIdx.x;
    const int b   = idx >> 14;           // / MEMU
    const float g  = ws[WS_G + b];
    const float se = ws[WS_SUMEXP + b];
    float wc = ereg[idx] / se;
    wi[idx] = fmaf(g, wc, (1.f - g) * prev_w[idx]);
}

// ---------------------------------------------------------------------------
// Kernel 4: circular 3-tap shift + sharpen + partition-sum of sharp.
// Each 256-thread block lies entirely inside one batch (256 | 16384).
// ---------------------------------------------------------------------------
__global__ void sharpen_kernel(const float* __restrict__ wi,
                               const float* __restrict__ ws,
                               float* __restrict__ sharp_out,   // d_out w-region
                               float* __restrict__ sumsh)
{
    __shared__ float red[256];
    const int t   = threadIdx.x;
    const int idx = blockIdx.x * 256 + t;
    const int b   = idx >> 14;
    const int m   = idx & (MEMU - 1);
    const int mm1 = (m == 0) ? MEMU - 1 : m - 1;
    const int mp1 = (m == MEMU - 1) ? 0 : m + 1;
    const float* wib = wi + (size_t)b * MEMU;
    const float s0 = ws[WS_S + 3 * b + 0];
    const float s1 = ws[WS_S + 3 * b + 1];
    const float s2 = ws[WS_S + 3 * b + 2];
    float sh = s0 * wib[mm1] + s1 * wib[m] + s2 * wib[mp1];
    float sp = powf(sh, ws[WS_Y + b]);   // sh >= 0, y >= 1
    sharp_out[idx] = sp;
    red[t] = sp;
    __syncthreads();
    for (int s = 128; s > 0; s >>= 1) { if (t < s) red[t] += red[t + s]; __syncthreads(); }
    if (t == 0) atomicAdd(&sumsh[b], red[0]);
}

// ---------------------------------------------------------------------------
// Kernel 5: normalize w in place, then second full pass over memory:
// data[b,:] += sum_m w[b,m] * mem[b,m,:]  (coalesced per-column accumulation).
// grid = B*32 blocks, each handles 512 memory rows.
// ---------------------------------------------------------------------------
#define CHROWS 512
__global__ void read_kernel(const float* __restrict__ mem,
                            const float* __restrict__ ws,
                            float* __restrict__ wreg,    // in: sharp, out: final w
                            float* __restrict__ data)    // d_out data region
{
    __shared__ float wl[CHROWS];
    __shared__ float part[256];
    const int t  = threadIdx.x;
    const int b  = blockIdx.x >> 5;
    const int m0 = (blockIdx.x & 31) * CHROWS;
    const float inv = 1.f / (ws[WS_SUMSH + b] + 1e-16f);

    for (int i = t; i < CHROWS; i += 256) {
        float v = wreg[(size_t)b * MEMU + m0 + i] * inv;
        wreg[(size_t)b * MEMU + m0 + i] = v;     // final w output
        wl[i] = v;
    }
    __syncthreads();

    const int col = t & 63, slice = t >> 6;      // 4 slices of 128 rows
    const float* mb = mem + ((size_t)b * MEMU + m0) * KEY;
    float acc = 0.f;
    for (int j = 0; j < CHROWS / 4; ++j) {
        int m = slice * (CHROWS / 4) + j;
        acc = fmaf(wl[m], mb[(size_t)m * KEY + col], acc);
    }
    part[t] = acc;
    __syncthreads();
    if (t < KEY) {
        float tot = part[t] + part[t + 64] + part[t + 128] + part[t + 192];
        atomicAdd(&data[b * KEY + t], tot);
    }
}

// ---------------------------------------------------------------------------
extern "C" void kernel_launch(void* const* d_in, const int* in_sizes, int n_in,
                              void* d_out, int out_size, void* d_ws, size_t ws_size,
                              hipStream_t stream)
{
    const float* cs     = (const float*)d_in[0];
    const float* prev_w = (const float*)d_in[1];
    const float* mem    = (const float*)d_in[2];
    const float* key_W  = (const float*)d_in[3];
    const float* key_b  = (const float*)d_in[4];
    const float* ks_W   = (const float*)d_in[5];
    const float* ks_b   = (const float*)d_in[6];
    const float* ig_W   = (const float*)d_in[7];
    const float* ig_b   = (const float*)d_in[8];
    const float* sw_W   = (const float*)d_in[9];
    const float* sw_b   = (const float*)d_in[10];
    const float* sf_W   = (const float*)d_in[11];
    const float* sf_b   = (const float*)d_in[12];
    // d_in[13], d_in[14] (wd_W, wd_b) unused by the reference outputs.

    float* ws   = (float*)d_ws;
    float* out  = (float*)d_out;
    float* wreg = out;                            // (B, MEMU) weights output
    float* data = out + (size_t)B * MEMU;         // (B, KEY) read-mode output

    head_params<<<B, 256, 0, stream>>>(cs, key_W, key_b, ks_W, ks_b, ig_W, ig_b,
                                       sw_W, sw_b, sf_W, sf_b, ws, out);
    dots_kernel<<<B * (MEMU / ROWS), 256, 0, stream>>>(mem, ws, wreg, ws + WS_SUMEXP);
    interp_kernel<<<(B * MEMU) / 256, 256, 0, stream>>>(prev_w, ws, wreg, ws + WS_BUF0);
    sharpen_kernel<<<(B * MEMU) / 256, 256, 0, stream>>>(ws + WS_BUF0, ws, wreg, ws + WS_SUMSH);
    read_kernel<<<B * 32, 256, 0, stream>>>(mem, ws, wreg, data);
}